// TorchRCModel3R1C_54288386622035
// MI455X (gfx1250) — compile-verified
//
#include <hip/hip_runtime.h>
#include <math.h>

// ---------------------------------------------------------------------------
// Affine scan  T[i] = a*T[i-1] + bb[i]  (T[-1] = 0), bb[0] = T_init,
// bb[i] = DT*(x[i-1,0]*invReq + x[i-1,1] + x[i-1,2] + x[i-1,3])/C_air.
// 3-pass chunked scan; 16x16 intra-tile scans done with v_wmma_f32_16x16x4_f32
// against the constant lower-triangular power matrix L[i][k] = a^(i-k).
// Per-wave LDS tiles; wave-local DScnt waits instead of workgroup barriers.
// ---------------------------------------------------------------------------

typedef float v2f __attribute__((ext_vector_type(2)));
typedef float v8f __attribute__((ext_vector_type(8)));

#define LCHUNK 4096          // elements per chunk (one wave per chunk)
#define TILES_PER_CHUNK 16   // 4096 / 256
#define WPB 8                // waves per 256-thread block
#define BSTR 20              // LDS stride (floats) for bb tile rows
#define DSTR 17              // LDS stride (floats) for D tile rows

struct RCParams {
  float Tinit, invReq, bcoef, a;
};

__device__ __forceinline__ RCParams rc_params(const float* pT, const float* pRe,
                                              const float* pRv, const float* pC) {
  RCParams p;
  p.Tinit = pT[0];
  float sRe = 1.0f / (1.0f + expf(-pRe[0]));
  float sRv = 1.0f / (1.0f + expf(-pRv[0]));
  float sC  = 1.0f / (1.0f + expf(-pC[0]));
  float Renv  = 0.001f + sRe * (0.1f - 0.001f);
  float Rvent = 0.001f + sRv * (0.1f - 0.001f);
  float Cair  = 100000.0f + sC * (10000000.0f - 100000.0f);
  p.invReq = 1.0f / Renv + 1.0f / Rvent;
  p.a      = 1.0f - (300.0f * p.invReq) / Cair;
  p.bcoef  = 300.0f / Cair;
  return p;
}

// a^e for e in [0,15] via 4-bit decomposition (a2=a^2, a4=a^4, a8=a^8)
__device__ __forceinline__ float powA4(float a, float a2, float a4, float a8, int e) {
  float p = 1.0f;
  if (e & 1) p *= a;
  if (e & 2) p *= a2;
  if (e & 4) p *= a4;
  if (e & 8) p *= a8;
  return p;
}

// Wave-local LDS RAW fence: all LDS regions are per-wave, so cross-lane
// visibility only requires DScnt==0 (LDS ops of one wave stay in order).
// Cheaper than s_barrier_signal/wait across the 8 independent waves.
__device__ __forceinline__ void wave_lds_sync() {
  asm volatile("s_wait_dscnt 0x0" ::: "memory");
}

// ---- Kernel 1: per-chunk carry C_c = sum_j a^(L-1-j) * bb[c*L + j] ----------
__global__ __launch_bounds__(256) void rc_k1_carry(const float4* __restrict__ inp,
                                                   const float* pT, const float* pRe,
                                                   const float* pRv, const float* pC,
                                                   float* __restrict__ carry) {
  __shared__ float s_bb[WPB][16 * BSTR];
  __shared__ float s_d [WPB][16 * DSTR];
  __shared__ float s_e [WPB][16];

  const int lane = threadIdx.x & 31;
  const int wv   = threadIdx.x >> 5;
  const int chunk = blockIdx.x * WPB + wv;
  const int base  = chunk * LCHUNK;

  RCParams P = rc_params(pT, pRe, pRv, pC);
  const float a = P.a;
  const float a2 = a * a, a4 = a2 * a2, a8 = a4 * a4;
  const float A16 = a8 * a8;
  const float A16_2 = A16 * A16, A16_4 = A16_2 * A16_2, A16_8 = A16_4 * A16_4;
  const float A256 = A16_8 * A16_8;

  // Constant B operand = L^T :  B[k][i] = a^(i-k) for k<=i else 0
  const int iN    = lane & 15;
  const int khalf = (lane >> 4) * 2;
  v2f bconst[4];
#pragma unroll
  for (int j = 0; j < 4; ++j) {
    int k0 = 4 * j + khalf;
    bconst[j].x = (iN >= k0)     ? powA4(a, a2, a4, a8, iN - k0)     : 0.0f;
    bconst[j].y = (iN >= k0 + 1) ? powA4(a, a2, a4, a8, iN - k0 - 1) : 0.0f;
  }

  float accv = 0.0f;
  for (int t = 0; t < TILES_PER_CHUNK; ++t) {
    const int tbase = base + t * 256;
    // Phase A: compute bb for 256 elements (coalesced float4 row loads)
#pragma unroll
    for (int u = 0; u < 8; ++u) {
      int el = lane + 32 * u;
      int e  = tbase + el;
      float bb;
      if (e == 0) {
        bb = P.Tinit;
      } else {
        float4 x = inp[e - 1];
        bb = P.bcoef * (x.x * P.invReq + x.y + x.z + x.w);
      }
      s_bb[wv][(el >> 4) * BSTR + (el & 15)] = bb;
    }
    wave_lds_sync();
    // Phase B: D = A(data 16x16) x L^T  via 4 chained 16x16x4 WMMAs
    v8f acc = {0.f, 0.f, 0.f, 0.f, 0.f, 0.f, 0.f, 0.f};
#pragma unroll
    for (int j = 0; j < 4; ++j) {
      int k0 = 4 * j + khalf;
      v2f av;
      av.x = s_bb[wv][(lane & 15) * BSTR + k0];
      av.y = s_bb[wv][(lane & 15) * BSTR + k0 + 1];
      acc = __builtin_amdgcn_wmma_f32_16x16x4_f32(false, av, false, bconst[j],
                                                  (short)0, acc, false, false);
    }
    // Phase C: stash D (only column 15 = segment carries is needed here)
#pragma unroll
    for (int r = 0; r < 8; ++r)
      s_d[wv][(r + 8 * (lane >> 4)) * DSTR + (lane & 15)] = acc[r];
    wave_lds_sync();
    // Phase D: weighted segment carries -> tile aggregate S_t
    if (lane < 16) {
      float c = s_d[wv][lane * DSTR + 15];
      s_e[wv][lane] = powA4(A16, A16_2, A16_4, A16_8, 15 - lane) * c;
    }
    wave_lds_sync();
    float S = 0.0f;
#pragma unroll
    for (int r = 0; r < 16; ++r) S += s_e[wv][r];
    accv = A256 * accv + S;
  }
  if (lane == 0) carry[chunk] = accv;
}

// ---- Kernel 2: spine — chunk start values (affine scan, ratio a^LCHUNK) ----
__global__ __launch_bounds__(256) void rc_k2_spine(const float* __restrict__ carry,
                                                   float* __restrict__ startv,
                                                   const float* pT, const float* pRe,
                                                   const float* pRv, const float* pC,
                                                   int nchunks) {
  __shared__ float sc[256];
  const int t = threadIdx.x;
  RCParams P = rc_params(pT, pRe, pRv, pC);
  float a = P.a;
  float p2 = a * a, p4 = p2 * p2, p8 = p4 * p4, p16 = p8 * p8;
  float p32 = p16 * p16, p64 = p32 * p32, p128 = p64 * p64, p256 = p128 * p128;
  float p512 = p256 * p256, p1024 = p512 * p512, p2048 = p1024 * p1024;
  const float AL = p2048 * p2048;  // a^4096
  const int PER = nchunks >> 8;    // chunks per thread (8 for N=2^23)
  // ALP = AL^PER
  float ALP = 1.0f;
  for (int i = 0; i < PER; ++i) ALP *= AL;

  const int c0 = t * PER;
  float r = 0.0f;
  for (int j = 0; j < PER; ++j) r = AL * r + carry[c0 + j];
  sc[t] = r;
  __syncthreads();
  float m = ALP;
  for (int d = 1; d < 256; d <<= 1) {
    float add = (t >= d) ? sc[t - d] : 0.0f;
    __syncthreads();
    sc[t] = sc[t] + m * add;
    __syncthreads();
    m = m * m;
  }
  float run = (t > 0) ? sc[t - 1] : 0.0f;
  for (int j = 0; j < PER; ++j) {
    startv[c0 + j] = run;
    run = AL * run + carry[c0 + j];
  }
}

// ---- Kernel 3: replay chunk with known start, write outputs ----------------
__global__ __launch_bounds__(256) void rc_k3_write(const float4* __restrict__ inp,
                                                   const float* pT, const float* pRe,
                                                   const float* pRv, const float* pC,
                                                   const float* __restrict__ startv,
                                                   float* __restrict__ out) {
  __shared__ float s_bb[WPB][16 * BSTR];
  __shared__ float s_d [WPB][16 * DSTR];
  __shared__ float s_e [WPB][16];

  const int lane = threadIdx.x & 31;
  const int wv   = threadIdx.x >> 5;
  const int chunk = blockIdx.x * WPB + wv;
  const int base  = chunk * LCHUNK;

  RCParams P = rc_params(pT, pRe, pRv, pC);
  const float a = P.a;
  const float a2 = a * a, a4 = a2 * a2, a8 = a4 * a4;
  const float A16 = a8 * a8;
  const float A16_2 = A16 * A16, A16_4 = A16_2 * A16_2, A16_8 = A16_4 * A16_4;

  const int iN    = lane & 15;
  const int khalf = (lane >> 4) * 2;
  v2f bconst[4];
#pragma unroll
  for (int j = 0; j < 4; ++j) {
    int k0 = 4 * j + khalf;
    bconst[j].x = (iN >= k0)     ? powA4(a, a2, a4, a8, iN - k0)     : 0.0f;
    bconst[j].y = (iN >= k0 + 1) ? powA4(a, a2, a4, a8, iN - k0 - 1) : 0.0f;
  }
  const float apow_i = powA4(a, a2, a4, a8, iN) * a;  // a^(i+1), i = lane&15

  float W = startv[chunk];  // scan value just before this chunk
  for (int t = 0; t < TILES_PER_CHUNK; ++t) {
    const int tbase = base + t * 256;
#pragma unroll
    for (int u = 0; u < 8; ++u) {
      int el = lane + 32 * u;
      int e  = tbase + el;
      float bb;
      if (e == 0) {
        bb = P.Tinit;
      } else {
        float4 x = inp[e - 1];
        bb = P.bcoef * (x.x * P.invReq + x.y + x.z + x.w);
      }
      s_bb[wv][(el >> 4) * BSTR + (el & 15)] = bb;
    }
    wave_lds_sync();
    v8f acc = {0.f, 0.f, 0.f, 0.f, 0.f, 0.f, 0.f, 0.f};
#pragma unroll
    for (int j = 0; j < 4; ++j) {
      int k0 = 4 * j + khalf;
      v2f av;
      av.x = s_bb[wv][(lane & 15) * BSTR + k0];
      av.y = s_bb[wv][(lane & 15) * BSTR + k0 + 1];
      acc = __builtin_amdgcn_wmma_f32_16x16x4_f32(false, av, false, bconst[j],
                                                  (short)0, acc, false, false);
    }
#pragma unroll
    for (int r = 0; r < 8; ++r)
      s_d[wv][(r + 8 * (lane >> 4)) * DSTR + (lane & 15)] = acc[r];
    wave_lds_sync();
    // E_s = a^(16s)*W + sum_{r<s} a^(16(s-1-r)) * carry_r
    if (lane < 16) {
      float E = powA4(A16, A16_2, A16_4, A16_8, lane) * W;
      float p = 1.0f;
      for (int r = lane - 1; r >= 0; --r) {
        E += p * s_d[wv][r * DSTR + 15];
        p *= A16;
      }
      s_e[wv][lane] = E;
    }
    wave_lds_sync();
    float Wn = A16 * s_e[wv][15] + s_d[wv][15 * DSTR + 15];
    // out[tbase + s*16 + i] = D[s][i] + a^(i+1) * E_s   (coalesced stores)
#pragma unroll
    for (int u = 0; u < 8; ++u) {
      int s2 = (lane >> 4) + 2 * u;
      float val = s_d[wv][s2 * DSTR + (lane & 15)] + apow_i * s_e[wv][s2];
      out[tbase + 32 * u + lane] = val;
    }
    W = Wn;
  }
}

extern "C" void kernel_launch(void* const* d_in, const int* in_sizes, int n_in,
                              void* d_out, int out_size, void* d_ws, size_t ws_size,
                              hipStream_t stream) {
  (void)in_sizes; (void)n_in; (void)ws_size;
  const float*  pT   = (const float*)d_in[0];   // T_init (scalar)
  const float4* inp  = (const float4*)d_in[1];  // inputs (N,4)
  const float*  pRe  = (const float*)d_in[2];   // w_R_env
  // d_in[3] = w_R_int (unused by reference)
  const float*  pRv  = (const float*)d_in[4];   // w_R_vent
  const float*  pC   = (const float*)d_in[5];   // w_C_air
  float* out = (float*)d_out;

  const int N = out_size;              // 8388608 = 2^23
  const int nchunks = N / LCHUNK;      // 2048
  float* carry  = (float*)d_ws;        // [nchunks]
  float* startv = carry + nchunks;     // [nchunks]

  dim3 blk(256);
  rc_k1_carry<<<nchunks / WPB, blk, 0, stream>>>(inp, pT, pRe, pRv, pC, carry);
  rc_k2_spine<<<1, blk, 0, stream>>>(carry, startv, pT, pRe, pRv, pC, nchunks);
  rc_k3_write<<<nchunks / WPB, blk, 0, stream>>>(inp, pT, pRe, pRv, pC, startv, out);
}